// Token_54674933678383
// MI455X (gfx1250) — compile-verified
//
#include <hip/hip_runtime.h>

// CDNA5 / gfx1250, wave32. One wave handles 3 samples (15 of 16 WMMA rows).
// Gram matrix via chained V_WMMA_F32_16X16X4_F32 (K=64 in 16 steps, exact fp32).
// B matrix (zero-embedded tokens) padded to 16 zero rows in LDS so all fragment
// loads are unconditional; A/B fragments preloaded so the 16 WMMAs issue
// back-to-back.

typedef float v2f __attribute__((ext_vector_type(2)));
typedef float v8f __attribute__((ext_vector_type(8)));

#define NSAMP 375
#define NCH   5
#define NF    64
#define ND    13
#define SPB   3          // samples per block (3*5 = 15 rows of the 16-row WMMA tile)

__device__ __constant__ int kStart[5] = {0, 12, 25, 38, 51};

__global__ __launch_bounds__(32)
void token_assign_kernel(const float* __restrict__ pic,
                         const float* __restrict__ t0,
                         const float* __restrict__ t1,
                         const float* __restrict__ t2,
                         const float* __restrict__ t3,
                         const float* __restrict__ t4,
                         float* __restrict__ out)
{
    __shared__ float Tf[16][NF];    // tokens embedded in zero-padded rows; rows 5..15 = 0
    __shared__ float Dls[16][16];   // cost matrix, row m = 5*sample + channel
    __shared__ int   Als[16];       // assignment per (sample, channel)

    const int lane = (int)threadIdx.x;
    const int n0   = (int)blockIdx.x * SPB;

    // ---- build embedded token matrix (16 rows, rows >=5 all zero) -------
    {
        const float* toks[5] = {t0, t1, t2, t3, t4};
        for (int i = lane; i < 16 * NF; i += 32) {
            const int s = i >> 6, f = i & 63;
            float v = 0.0f;
            if (s < 5) {
                const int d = f - kStart[s];
                if (d >= 0 && d < ND) v = toks[s][d];
            }
            Tf[s][f] = v;
        }
    }
    __syncthreads();

    // ---- fragment addressing (A 16x4 f32 / B 4x16 f32 layouts) ----------
    const int col   = lane & 15;                 // B/C/D column (N) this lane feeds
    const int m     = lane & 15;                 // A-matrix row this lane supplies
    const int mrow  = (m < 15) ? m : 14;         // clamp row 15 (unused) in bounds
    const int khalf = (lane >> 4) << 1;          // lanes 0-15: K+{0,1}; 16-31: K+{2,3}
    const float* rp = pic + (size_t)(n0 * NCH + mrow) * NF;

    // ---- preload all fragments, fold in row/col norms --------------------
    v2f av[16], bv[16];
    #pragma unroll
    for (int t = 0; t < 16; ++t) {               // 16x global_load_b64, pipelined
        av[t] = *(const v2f*)(rp + 4 * t + khalf);
    }
    #pragma unroll
    for (int t = 0; t < 16; ++t) {               // 16x ds_load_b64, unconditional
        bv[t] = *(const v2f*)(&Tf[col][4 * t + khalf]);
    }

    float sq = 0.0f, tb = 0.0f;
    #pragma unroll
    for (int t = 0; t < 16; ++t) {
        sq += av[t].x * av[t].x + av[t].y * av[t].y;   // partial ||P[row m]||^2
        tb += bv[t].x * bv[t].x + bv[t].y * bv[t].y;   // partial ||T[col]||^2
    }

    // ---- 16 back-to-back fp32 WMMAs: D[m,s] = sum_f P[m,f] * Tf[s,f] ----
    v8f acc = {};
    #pragma unroll
    for (int t = 0; t < 16; ++t) {
        acc = __builtin_amdgcn_wmma_f32_16x16x4_f32(
                  false, av[t], false, bv[t], (short)0, acc, false, false);
    }

    // combine half-wave partials (row m held by lanes m and m+16)
    sq += __shfl_xor(sq, 16, 32);                // full_sq[m] on lanes m, m+16
    const float tn = tb + __shfl_xor(tb, 16, 32);// ||T[col]||^2 (0 for col>=5)

    // ---- Dmat = full_sq + ||T||^2 - 2*dot; park in LDS (unconditional) --
    const int half8 = (lane < 16) ? 0 : 8;
    #pragma unroll
    for (int v = 0; v < 8; ++v) {
        const int   mr = v + half8;              // D-matrix row held in acc[v]
        const float fs = __shfl(sq, mr, 32);     // broadcast full_sq[mr]
        Dls[mr][col] = fs + tn - 2.0f * acc[v];
    }
    __syncthreads();

    // ---- greedy bipartite matching, one lane per sample -----------------
    if (lane < SPB) {
        float dm[25];
        #pragma unroll
        for (int r = 0; r < 5; ++r)
            #pragma unroll
            for (int c = 0; c < 5; ++c)
                dm[r * 5 + c] = Dls[lane * 5 + r][c];

        unsigned rowUsed = 0, colUsed = 0;
        int assign[5];
        for (int it = 0; it < 5; ++it) {
            float best = __builtin_inff();
            int   bi   = 0;
            for (int idx = 0; idx < 25; ++idx) {  // flat order => first-occurrence argmin
                const int r = idx / 5, c = idx % 5;
                const bool ok = !((rowUsed >> r) & 1u) && !((colUsed >> c) & 1u);
                if (ok && dm[idx] < best) { best = dm[idx]; bi = idx; }
            }
            const int r = bi / 5, c = bi % 5;
            assign[r] = c;
            rowUsed |= 1u << r;
            colUsed |= 1u << c;
        }
        #pragma unroll
        for (int r = 0; r < 5; ++r) Als[lane * 5 + r] = assign[r];
    }
    __syncthreads();

    // ---- scatter output: out[n,c,start[c]+d] = T[assign[n,c], d] --------
    float* ob = out + (size_t)n0 * NCH * NF;
    #pragma unroll
    for (int e = lane; e < SPB * NCH * NF; e += 32) {
        const int j  = e / (NCH * NF);
        const int cf = e - j * (NCH * NF);
        const int c  = cf >> 6;
        const int f  = cf & 63;
        const int d  = f - kStart[c];
        float val = 0.0f;
        if (d >= 0 && d < ND) {
            const int s = Als[j * NCH + c];
            val = Tf[s][kStart[s] + d];
        }
        ob[e] = val;
    }
}

extern "C" void kernel_launch(void* const* d_in, const int* in_sizes, int n_in,
                              void* d_out, int out_size, void* d_ws, size_t ws_size,
                              hipStream_t stream) {
    (void)in_sizes; (void)n_in; (void)out_size; (void)d_ws; (void)ws_size;
    const float* pic = (const float*)d_in[0];
    const float* t0  = (const float*)d_in[1];
    const float* t1  = (const float*)d_in[2];
    const float* t2  = (const float*)d_in[3];
    const float* t3  = (const float*)d_in[4];
    const float* t4  = (const float*)d_in[5];
    float* out = (float*)d_out;

    const int grid = NSAMP / SPB;   // 375 / 3 = 125 single-wave workgroups
    token_assign_kernel<<<grid, 32, 0, stream>>>(pic, t0, t1, t2, t3, t4, out);
}